// MetabolismProcessor_61899068670355
// MI455X (gfx1250) — compile-verified
//
#include <hip/hip_runtime.h>
#include <math.h>

#define DD 256
static constexpr int kNMet  = 2534;
static constexpr int kNRxn  = 4881;
static constexpr int kNGene = 6607;
static constexpr int kE     = 262144;
static constexpr int kP     = 131072;
// GEMM row padding: multiple of 128 (8 waves * 16-row tiles per block)
static constexpr int kNMetP = 2560;   // ceil(2534/128)*128
static constexpr int kNRxnP = 4992;   // ceil(4881/128)*128

typedef __attribute__((ext_vector_type(2))) float v2f;
typedef __attribute__((ext_vector_type(8))) float v8f;

__device__ __forceinline__ float waveSum(float v) {
#pragma unroll
  for (int off = 16; off > 0; off >>= 1) v += __shfl_down(v, off, 32);
  return __shfl(v, 0, 32);
}

// ---------------- generic fill ----------------
__global__ void k_fill(float* __restrict__ p, float v, int n) {
  int i = blockIdx.x * blockDim.x + threadIdx.x;
  if (i < n) p[i] = v;
}

// ---------------- histogram counts: cnt[idx[i]] += 1 ----------------
__global__ void k_count(const int* __restrict__ idx, float* __restrict__ cnt, int n) {
  int i = blockIdx.x * blockDim.x + threadIdx.x;
  if (i < n) atomicAdd(&cnt[idx[i]], 1.0f);
}

// ---------------- embedding max-norm renorm (one wave per row) ----------------
__global__ void k_renorm(const float* __restrict__ emb, float* __restrict__ out, int rows) {
  int gw = (blockIdx.x * blockDim.x + threadIdx.x) >> 5;
  int lane = threadIdx.x & 31;
  if (gw >= rows) return;
  const float* r = emb + (size_t)gw * DD;
  float ss = 0.f;
#pragma unroll
  for (int j = 0; j < DD / 32; ++j) { float v = r[lane + 32 * j]; ss += v * v; }
  ss = waveSum(ss);
  float sc = fminf(1.0f, 1.0f / (sqrtf(ss) + 1e-12f));
  float* o = out + (size_t)gw * DD;
#pragma unroll
  for (int j = 0; j < DD / 32; ++j) o[lane + 32 * j] = r[lane + 32 * j] * sc;
}

// ---------------- fp32 WMMA GEMM: C[Mpad,N] = A[Mpad,K] @ B[K,N] ----------------
// Block = 8 waves = 8 M-tiles x one 64-col stripe. B stripe staged in LDS
// (transposed, K chunked by 128). One wave -> 16x64 output = 4 accumulators.
// Mpad is a multiple of 128 so there is NO predication anywhere.
#define KCH 128          // K chunk staged in LDS
#define BSTR 130         // LDS stride (floats) per column: conflict-free b64 reads
__global__ void k_gemm_wmma(const float* __restrict__ A, const float* __restrict__ B,
                            float* __restrict__ C, int Mpad, int N, int K) {
  __shared__ float lb[64 * BSTR];            // 33.3 KB
  int nStripes = N >> 6;                     // N/64
  int bm = blockIdx.x / nStripes;
  int colBase = (blockIdx.x % nStripes) << 6;
  int wid  = threadIdx.x >> 5;
  int lane = threadIdx.x & 31;
  int half = lane >> 4;                      // K sub-pair select
  int l16  = lane & 15;

  int rowA = (bm * 8 + wid) * 16 + l16;
  const float* Arow = A + (size_t)rowA * K + half * 2;

  v8f acc0 = {}, acc1 = {}, acc2 = {}, acc3 = {};

  for (int kc = 0; kc < K; kc += KCH) {
    __syncthreads();
    // cooperative stage: B[kc..kc+127][colBase..colBase+63] -> lb[c*BSTR + k]
    for (int idx = threadIdx.x; idx < KCH * 16; idx += 256) {
      int kl = idx >> 4;                     // 0..127
      int c4 = (idx & 15) << 2;              // 0..60 step 4
      float4 v = *(const float4*)(B + (size_t)(kc + kl) * N + colBase + c4);
      lb[(c4 + 0) * BSTR + kl] = v.x;
      lb[(c4 + 1) * BSTR + kl] = v.y;
      lb[(c4 + 2) * BSTR + kl] = v.z;
      lb[(c4 + 3) * BSTR + kl] = v.w;
    }
    __syncthreads();
#pragma unroll 8
    for (int k = 0; k < KCH; k += 4) {
      int kk = k + half * 2;                 // even -> 8B aligned
      v2f a  = *(const v2f*)(Arow + kc + k); // global_load_b64
      v2f b0 = *(const v2f*)&lb[(l16 +  0) * BSTR + kk];   // ds_load_b64
      v2f b1 = *(const v2f*)&lb[(l16 + 16) * BSTR + kk];
      v2f b2 = *(const v2f*)&lb[(l16 + 32) * BSTR + kk];
      v2f b3 = *(const v2f*)&lb[(l16 + 48) * BSTR + kk];
      acc0 = __builtin_amdgcn_wmma_f32_16x16x4_f32(false, a, false, b0, (short)0, acc0, false, false);
      acc1 = __builtin_amdgcn_wmma_f32_16x16x4_f32(false, a, false, b1, (short)0, acc1, false, false);
      acc2 = __builtin_amdgcn_wmma_f32_16x16x4_f32(false, a, false, b2, (short)0, acc2, false, false);
      acc3 = __builtin_amdgcn_wmma_f32_16x16x4_f32(false, a, false, b3, (short)0, acc3, false, false);
    }
  }

  // store 16x64 tile (all rows valid: Mpad padded)
  int rowBase = (bm * 8 + wid) * 16 + half * 8;
#pragma unroll
  for (int r = 0; r < 8; ++r) {
    float* crow = C + (size_t)(rowBase + r) * N + colBase + l16;
    crow[0]  = acc0[r];
    crow[16] = acc1[r];
    crow[32] = acc2[r];
    crow[48] = acc3[r];
  }
}

// ---------------- s[row] = dot(X[row,:], att[:]) (one wave per row) ----------------
__global__ void k_rowdot(const float* __restrict__ X, const float* __restrict__ att,
                         float* __restrict__ s, int rows) {
  int gw = (blockIdx.x * blockDim.x + threadIdx.x) >> 5;
  int lane = threadIdx.x & 31;
  if (gw >= rows) return;
  const float* r = X + (size_t)gw * DD;
  float acc = 0.f;
#pragma unroll
  for (int j = 0; j < DD / 32; ++j) acc += r[lane + 32 * j] * att[lane + 32 * j];
  acc = waveSum(acc);
  if (lane == 0) s[gw] = acc;
}

// ---------------- logit + segment max ----------------
__global__ void k_logit_max(const int* __restrict__ node, const int* __restrict__ edge,
                            const float* __restrict__ sm, const float* __restrict__ sr,
                            float* __restrict__ logits, float* __restrict__ segmax, int n) {
  int i = blockIdx.x * blockDim.x + threadIdx.x;
  if (i >= n) return;
  float x = sm[node[i]] + sr[edge[i]];
  float l = x > 0.f ? x : 0.2f * x;       // leaky_relu(0.2)
  logits[i] = l;
  atomicMax(&segmax[edge[i]], l);         // global_atomic_max_num_f32
}

// ---------------- exp + segment sum (in place on logits buffer) ----------------
__global__ void k_exp_sum(const int* __restrict__ edge, const float* __restrict__ segmax,
                          float* __restrict__ ex, float* __restrict__ ssum, int n) {
  int i = blockIdx.x * blockDim.x + threadIdx.x;
  if (i >= n) return;
  float e = expf(ex[i] - segmax[edge[i]]);
  ex[i] = e;
  atomicAdd(&ssum[edge[i]], e);
}

// ---------------- alpha*stoich (in place) ----------------
__global__ void k_alpha_w(const int* __restrict__ edge, const float* __restrict__ ssum,
                          const float* __restrict__ stoich, float* __restrict__ w, int n) {
  int i = blockIdx.x * blockDim.x + threadIdx.x;
  if (i >= n) return;
  w[i] = stoich[i] * (w[i] / (ssum[edge[i]] + 1e-16f));
}

// ---------------- acc[dst[i],:] += src[srcIdx[i],:] (block = one incidence) ----------------
__global__ void k_scat_add(const int* __restrict__ srcIdx, const int* __restrict__ dstIdx,
                           const float* __restrict__ src, float* __restrict__ acc, int n) {
  int i = blockIdx.x, d = threadIdx.x;
  int s = srcIdx[i], t = dstIdx[i];
  atomicAdd(&acc[(size_t)t * DD + d], src[(size_t)s * DD + d]);
}

// ---------------- acc[dst[i],:] += w[i]*src[srcIdx[i],:] ----------------
__global__ void k_scat_add_w(const int* __restrict__ srcIdx, const int* __restrict__ dstIdx,
                             const float* __restrict__ wt, const float* __restrict__ src,
                             float* __restrict__ acc, int n) {
  int i = blockIdx.x, d = threadIdx.x;
  int s = srcIdx[i], t = dstIdx[i];
  float wv = wt[i];
  atomicAdd(&acc[(size_t)t * DD + d], wv * src[(size_t)s * DD + d]);
}

// ---------------- x[r,:] /= max(cnt[r],1) ----------------
__global__ void k_rownorm_cnt(float* __restrict__ x, const float* __restrict__ cnt, int rows) {
  int r = blockIdx.x, d = threadIdx.x;
  x[(size_t)r * DD + d] /= fmaxf(cnt[r], 1.0f);
}

// ---------------- layer-0 epilogue: cur = tanh(out/max(Dc,1) + b) ----------------
__global__ void k_epi0(const float* __restrict__ outb, const float* __restrict__ Dc,
                       const float* __restrict__ bias, float* __restrict__ cur, int rows) {
  int r = blockIdx.x, d = threadIdx.x;
  float v = outb[(size_t)r * DD + d] / fmaxf(Dc[r], 1.0f) + bias[d];
  cur[(size_t)r * DD + d] = tanhf(v);
}

// ---------------- layer-1 epilogue fused: cur = LN(tanh(out/Dc + b) + cur) ----------------
__global__ void k_epi1_ln(const float* __restrict__ outb, const float* __restrict__ Dc,
                          const float* __restrict__ bias, const float* __restrict__ g,
                          const float* __restrict__ be, float* __restrict__ cur, int rows) {
  int gw = (blockIdx.x * blockDim.x + threadIdx.x) >> 5;
  int lane = threadIdx.x & 31;
  if (gw >= rows) return;
  float dcv = fmaxf(Dc[gw], 1.0f);
  const float* ob = outb + (size_t)gw * DD;
  float* cr = cur + (size_t)gw * DD;
  float z[DD / 32];
  float s = 0.f;
#pragma unroll
  for (int j = 0; j < DD / 32; ++j) {
    int d = lane + 32 * j;
    float t = tanhf(ob[d] / dcv + bias[d]);
    z[j] = t + cr[d];
    s += z[j];
  }
  float mu = waveSum(s) * (1.0f / DD);
  float v = 0.f;
#pragma unroll
  for (int j = 0; j < DD / 32; ++j) { float t = z[j] - mu; v += t * t; }
  float inv = rsqrtf(waveSum(v) * (1.0f / DD) + 1e-5f);
#pragma unroll
  for (int j = 0; j < DD / 32; ++j) {
    int d = lane + 32 * j;
    cr[d] = (z[j] - mu) * inv * g[d] + be[d];
  }
}

static inline int cdiv(int a, int b) { return (a + b - 1) / b; }

static void launch_gemm(const float* A, const float* B, float* C, int Mpad, int N, int K,
                        hipStream_t s) {
  int grid = (Mpad / 128) * (N / 64);
  k_gemm_wmma<<<grid, 256, 0, s>>>(A, B, C, Mpad, N, K);
}

extern "C" void kernel_launch(void* const* d_in, const int* in_sizes, int n_in,
                              void* d_out, int out_size, void* d_ws, size_t ws_size,
                              hipStream_t stream) {
  (void)in_sizes; (void)n_in; (void)out_size; (void)ws_size;
  const int*   he_node  = (const int*)d_in[0];
  const int*   he_edge  = (const int*)d_in[1];
  const float* stoich   = (const float*)d_in[2];
  const float* gene_x   = (const float*)d_in[3];
  const int*   rtg_rxn  = (const int*)d_in[4];
  const int*   rtg_gene = (const int*)d_in[5];
  const float* emb      = (const float*)d_in[6];
  const float* Ws[2]   = {(const float*)d_in[7],  (const float*)d_in[11]};
  const float* Wes[2]  = {(const float*)d_in[8],  (const float*)d_in[12]};
  const float* atts[2] = {(const float*)d_in[9],  (const float*)d_in[13]};
  const float* bs[2]   = {(const float*)d_in[10], (const float*)d_in[14]};
  const float* ln_g = (const float*)d_in[15];
  const float* ln_b = (const float*)d_in[16];
  float* out = (float*)d_out;

  // ---- workspace carve (floats, 64-elt aligned). GEMM operands row-padded. ----
  float* ws = (float*)d_ws;
  size_t off = 0;
  auto carve = [&](size_t n) { float* p = ws + off; off += (n + 63) & ~(size_t)63; return p; };
  float* met     = carve((size_t)kNMetP * DD);   // GEMM A (layer 0)
  float* cur     = carve((size_t)kNMetP * DD);   // GEMM A (layer 1)
  float* xp      = carve((size_t)kNMetP * DD);   // GEMM C
  float* rxn_emb = carve((size_t)kNRxnP * DD);   // GEMM A
  float* ep      = carve((size_t)kNRxnP * DD);   // GEMM C
  float* me      = carve((size_t)kNRxn * DD);
  float* outb    = carve((size_t)kNMet * DD);
  float* rf      = carve((size_t)kNRxn * DD);
  float* sm      = carve(kNMet);
  float* sr      = carve(kNRxn);
  float* segm    = carve(kNRxn);
  float* ssum    = carve(kNRxn);
  float* Bc      = carve(kNRxn);
  float* Dc      = carve(kNMet);
  float* rcnt    = carve(kNRxn);
  float* gcnt    = carve(kNGene);
  float* ebuf    = carve(kE);        // logits -> exp -> w, in place

  // ---- incidence/count histograms ----
  k_fill<<<cdiv(kNRxn, 256), 256, 0, stream>>>(Bc, 0.f, kNRxn);
  k_fill<<<cdiv(kNMet, 256), 256, 0, stream>>>(Dc, 0.f, kNMet);
  k_fill<<<cdiv(kNRxn, 256), 256, 0, stream>>>(rcnt, 0.f, kNRxn);
  k_fill<<<cdiv(kNGene, 256), 256, 0, stream>>>(gcnt, 0.f, kNGene);
  k_count<<<cdiv(kE, 256), 256, 0, stream>>>(he_edge, Bc, kE);
  k_count<<<cdiv(kE, 256), 256, 0, stream>>>(he_node, Dc, kE);
  k_count<<<cdiv(kP, 256), 256, 0, stream>>>(rtg_rxn, rcnt, kP);
  k_count<<<cdiv(kP, 256), 256, 0, stream>>>(rtg_gene, gcnt, kP);

  // ---- zero GEMM A-operand pad rows so padded tiles stay finite ----
  k_fill<<<cdiv((kNMetP - kNMet) * DD, 256), 256, 0, stream>>>(met + (size_t)kNMet * DD, 0.f,
                                                               (kNMetP - kNMet) * DD);
  k_fill<<<cdiv((kNMetP - kNMet) * DD, 256), 256, 0, stream>>>(cur + (size_t)kNMet * DD, 0.f,
                                                               (kNMetP - kNMet) * DD);
  k_fill<<<cdiv((kNRxnP - kNRxn) * DD, 256), 256, 0, stream>>>(rxn_emb + (size_t)kNRxn * DD, 0.f,
                                                               (kNRxnP - kNRxn) * DD);

  // ---- met = renorm(emb_table) ----
  k_renorm<<<cdiv(kNMet * 32, 256), 256, 0, stream>>>(emb, met, kNMet);

  // ---- rxn_emb = segment_mean(gene_x[rtg_gene], rtg_rxn) ----
  k_fill<<<cdiv(kNRxn * DD, 256), 256, 0, stream>>>(rxn_emb, 0.f, kNRxn * DD);
  k_scat_add<<<kP, DD, 0, stream>>>(rtg_gene, rtg_rxn, gene_x, rxn_emb, kP);
  k_rownorm_cnt<<<kNRxn, DD, 0, stream>>>(rxn_emb, rcnt, kNRxn);

  // ---- two hyperconv layers ----
  for (int l = 0; l < 2; ++l) {
    const float* X = (l == 0) ? met : cur;
    launch_gemm(X, Ws[l], xp, kNMetP, DD, DD, stream);          // xp = x @ W    (WMMA)
    launch_gemm(rxn_emb, Wes[l], ep, kNRxnP, DD, DD, stream);   // ep = eattr@We (WMMA)
    k_rowdot<<<cdiv(kNMet * 32, 256), 256, 0, stream>>>(xp, atts[l], sm, kNMet);
    k_rowdot<<<cdiv(kNRxn * 32, 256), 256, 0, stream>>>(ep, atts[l] + DD, sr, kNRxn);
    k_fill<<<cdiv(kNRxn, 256), 256, 0, stream>>>(segm, -3.0e38f, kNRxn);
    k_logit_max<<<cdiv(kE, 256), 256, 0, stream>>>(he_node, he_edge, sm, sr, ebuf, segm, kE);
    k_fill<<<cdiv(kNRxn, 256), 256, 0, stream>>>(ssum, 0.f, kNRxn);
    k_exp_sum<<<cdiv(kE, 256), 256, 0, stream>>>(he_edge, segm, ebuf, ssum, kE);
    k_alpha_w<<<cdiv(kE, 256), 256, 0, stream>>>(he_edge, ssum, stoich, ebuf, kE);
    // me = segsum(w * xp[node], edge) / max(Bc,1)
    k_fill<<<cdiv(kNRxn * DD, 256), 256, 0, stream>>>(me, 0.f, kNRxn * DD);
    k_scat_add_w<<<kE, DD, 0, stream>>>(he_node, he_edge, ebuf, xp, me, kE);
    k_rownorm_cnt<<<kNRxn, DD, 0, stream>>>(me, Bc, kNRxn);
    // out = segsum(w * me[edge], node) / max(Dc,1)
    k_fill<<<cdiv(kNMet * DD, 256), 256, 0, stream>>>(outb, 0.f, kNMet * DD);
    k_scat_add_w<<<kE, DD, 0, stream>>>(he_edge, he_node, ebuf, me, outb, kE);
    if (l == 0)
      k_epi0<<<kNMet, DD, 0, stream>>>(outb, Dc, bs[l], cur, kNMet);
    else
      k_epi1_ln<<<cdiv(kNMet * 32, 256), 256, 0, stream>>>(outb, Dc, bs[l], ln_g, ln_b,
                                                           cur, kNMet);
  }

  // ---- rxn_final = segment_mean(cur[he_node], he_edge) ----
  k_fill<<<cdiv(kNRxn * DD, 256), 256, 0, stream>>>(rf, 0.f, kNRxn * DD);
  k_scat_add<<<kE, DD, 0, stream>>>(he_node, he_edge, cur, rf, kE);
  k_rownorm_cnt<<<kNRxn, DD, 0, stream>>>(rf, Bc, kNRxn);

  // ---- gene_emb = segment_mean(rxn_final[rtg_rxn], rtg_gene) -> d_out ----
  k_fill<<<cdiv(kNGene * DD, 256), 256, 0, stream>>>(out, 0.f, kNGene * DD);
  k_scat_add<<<kP, DD, 0, stream>>>(rtg_rxn, rtg_gene, rf, out, kP);
  k_rownorm_cnt<<<kNGene, DD, 0, stream>>>(out, gcnt, kNGene);
}